// CRF_37228776522018
// MI455X (gfx1250) — compile-verified
//
#include <hip/hip_runtime.h>
#include <cstdint>
#include <cstddef>

// Problem dims (fixed by the reference)
#define S_LEN 512
#define B_N   256
#define T_N   128
#define BT    (B_N * T_N)      // 32768 elements per time step
#define CHUNK 73               // 7 * 73 = 511 backpointer steps
#define NCHUNK 7

// ---------------------------------------------------------------------------
// CDNA5 async global->LDS copy (one f32 per lane) + async counter waits.
// the LDS address comes from the VDST VGPR; low 32 bits of a generic LDS pointer are the
// group-segment byte offset, which is exactly what the instruction wants.
// ---------------------------------------------------------------------------
__device__ __forceinline__ void async_ld_f32(void* lds_dst, const float* g) {
  uint32_t loff = (uint32_t)(uintptr_t)lds_dst;
  uint64_t ga   = (uint64_t)(uintptr_t)g;
  asm volatile("global_load_async_to_lds_b32 %0, %1, off"
               :: "v"(loff), "v"(ga) : "memory");
}

// ===========================================================================
// K1: forward Viterbi recurrence. One block per batch, one thread per tag.
// transitions column lives in 128 VGPRs per thread; v broadcast via LDS.
// ===========================================================================
__global__ __launch_bounds__(128) void viterbi_fwd(
    const float* __restrict__ feats, const float* __restrict__ trans,
    const float* __restrict__ start_t, const float* __restrict__ end_t,
    unsigned char* __restrict__ bp, int* __restrict__ best_last,
    float* __restrict__ scores, int* __restrict__ paths)
{
  __shared__ __align__(16) float stage[32 * 128];  // 16 KB staging for trans
  __shared__ __align__(16) float ring[4 * 128];    // async feats prefetch ring
  __shared__ __align__(16) float vbuf[2 * 128];    // double-buffered v

  const int cur  = threadIdx.x;          // tag handled by this thread
  const int b    = blockIdx.x;           // batch handled by this block
  const int base = b * T_N + cur;        // offset within one [B,T] slab

  // ---- load my transitions column trans[prev][cur] into registers --------
  float tcol[128];
#pragma unroll
  for (int cc = 0; cc < 4; ++cc) {       // 4 chunks of 32 rows each
#pragma unroll
    for (int k = 0; k < 32; ++k)         // coalesced global -> LDS
      stage[(k << 7) + cur] = trans[cc * 4096 + (k << 7) + cur];
    __syncthreads();
#pragma unroll
    for (int j = 0; j < 32; ++j)         // conflict-free LDS -> VGPR column
      tcol[cc * 32 + j] = stage[(j << 7) + cur];
    __syncthreads();
  }

  // ---- step 0: v = feats[0] + start_transitions ---------------------------
  float vnew = feats[base] + start_t[cur];
  vbuf[cur] = vnew;

  // prime the async prefetch ring with feats for steps 1..4
#pragma unroll
  for (int s = 1; s <= 4; ++s)
    async_ld_f32(&ring[((s & 3) << 7) + cur], feats + (size_t)s * BT + base);
  __syncthreads();

  // ---- main recurrence ----------------------------------------------------
  int pb = 0;
  for (int s = 1; s < S_LEN; ++s) {
    // exactly 4 asyncs in flight at all times -> waiting to <=3 completes step s
    asm volatile("s_wait_asynccnt 0x3" ::: "memory");
    float emit = ring[((s & 3) << 7) + cur];

    // deep L2 prefetch of future feats
    {
      int pf = s + 16; if (pf > 511) pf = 511;
      __builtin_prefetch(feats + (size_t)pf * BT + base, 0, 1);
    }

    const float4* vb = (const float4*)(vbuf + (pb << 7));
    float b0 = -3.402823466e38f, b1 = b0, b2 = b0, b3 = b0;
    int a0 = 0, a1 = 1, a2 = 2, a3 = 3;
#pragma unroll
    for (int g = 0; g < 32; ++g) {
      float4 vv = vb[g];                       // uniform ds_load_b128 broadcast
      float c0 = vv.x + tcol[4 * g + 0];
      float c1 = vv.y + tcol[4 * g + 1];
      float c2 = vv.z + tcol[4 * g + 2];
      float c3 = vv.w + tcol[4 * g + 3];
      if (c0 > b0) { b0 = c0; a0 = 4 * g + 0; }   // 4 independent chains (ILP)
      if (c1 > b1) { b1 = c1; a1 = 4 * g + 1; }
      if (c2 > b2) { b2 = c2; a2 = 4 * g + 2; }
      if (c3 > b3) { b3 = c3; a3 = 4 * g + 3; }
    }
    // merge with argmax-first tie-break (smallest prev wins on equality)
    float best = b0; int arg = a0;
    if (b1 > best || (b1 == best && a1 < arg)) { best = b1; arg = a1; }
    if (b2 > best || (b2 == best && a2 < arg)) { best = b2; arg = a2; }
    if (b3 > best || (b3 == best && a3 < arg)) { best = b3; arg = a3; }

    vnew = best + emit;
    int cb = pb ^ 1;
    vbuf[(cb << 7) + cur] = vnew;
    bp[(size_t)(s - 1) * BT + base] = (unsigned char)arg;
    __syncthreads();   // waits this wave's DS reads, then barrier

    // keep in-flight async count constant (tail issues are harmless dummies
    // into the slot that was just consumed this step)
    int ps = s + 4; if (ps > 511) ps = 507;
    async_ld_f32(&ring[((s & 3) << 7) + cur], feats + (size_t)ps * BT + base);
    pb = cb;
  }

  // drain asyncs before reusing the ring LDS for the reduction
  asm volatile("s_wait_asynccnt 0x0" ::: "memory");
  __syncthreads();

  // ---- final argmax over tags (incl. end_transitions) ---------------------
  float vfin = vnew + end_t[cur];
  float* redv = ring;                 // reuse ring: 128 floats + 128 ints
  int*   redi = (int*)(ring + 128);
  redv[cur] = vfin;
  redi[cur] = cur;
  __syncthreads();
#pragma unroll
  for (int off = 64; off > 0; off >>= 1) {
    if (cur < off) {
      float av = redv[cur], bv = redv[cur + off];
      int   ai = redi[cur], bi = redi[cur + off];
      if (bv > av || (bv == av && bi < ai)) { redv[cur] = bv; redi[cur] = bi; }
    }
    __syncthreads();
  }
  if (cur == 0) {
    scores[b]    = redv[0];
    best_last[b] = redi[0];
    paths[(size_t)(S_LEN - 1) * B_N + b] = redi[0];  // paths[511] = best_last
  }
}

// ===========================================================================
// K2: compose each 73-step backpointer chunk into a tag->tag function F.
// block = (chunk, batch); thread t chases the chain starting from tag t.
// ===========================================================================
__global__ __launch_bounds__(128) void viterbi_compose(
    const unsigned char* __restrict__ bp, unsigned char* __restrict__ Fc)
{
  __shared__ uint32_t ch[CHUNK * 32];   // 73 rows x 128 bytes
  const int c   = blockIdx.x >> 8;
  const int b   = blockIdx.x & 255;
  const int sub = threadIdx.x >> 5;
  const int w   = threadIdx.x & 31;
  for (int k = 0; k < 19; ++k) {
    int r = sub + (k << 2);
    if (r < CHUNK) {
      const uint32_t* g =
          (const uint32_t*)(bp + (size_t)(c * CHUNK + r) * BT + (size_t)b * T_N);
      ch[r * 32 + w] = g[w];
    }
  }
  __syncthreads();
  const unsigned char* ch8 = (const unsigned char*)ch;
  int x = threadIdx.x;
  for (int i = CHUNK - 1; i >= 0; --i)
    x = ch8[(i << 7) + x];
  Fc[(size_t)(c * B_N + b) * T_N + threadIdx.x] = (unsigned char)x;
}

// ===========================================================================
// K3: walk the 7 chunk boundaries per batch (one thread per batch).
// ===========================================================================
__global__ void viterbi_boundary(const unsigned char* __restrict__ Fc,
                                 const int* __restrict__ best_last,
                                 int* __restrict__ btag)
{
  int b = threadIdx.x;
  int x = best_last[b];
  btag[NCHUNK * B_N + b] = x;
  for (int c = NCHUNK - 1; c >= 0; --c) {
    x = Fc[(size_t)(c * B_N + b) * T_N + x];
    btag[c * B_N + b] = x;
  }
}

// ===========================================================================
// K4: fill the interior of each chunk's path. Threads chase all 128
// hypotheses into LDS, then select the true one and write 73 path entries.
// ===========================================================================
__global__ __launch_bounds__(128) void viterbi_fill(
    const unsigned char* __restrict__ bp, const int* __restrict__ btag,
    int* __restrict__ paths)
{
  __shared__ uint32_t ch[CHUNK * 32];
  __shared__ unsigned char pos[CHUNK * 128];
  const int c   = blockIdx.x >> 8;
  const int b   = blockIdx.x & 255;
  const int sub = threadIdx.x >> 5;
  const int w   = threadIdx.x & 31;
  for (int k = 0; k < 19; ++k) {
    int r = sub + (k << 2);
    if (r < CHUNK) {
      const uint32_t* g =
          (const uint32_t*)(bp + (size_t)(c * CHUNK + r) * BT + (size_t)b * T_N);
      ch[r * 32 + w] = g[w];
    }
  }
  __syncthreads();
  const unsigned char* ch8 = (const unsigned char*)ch;
  int x = threadIdx.x;                      // hypothesis: tag at chunk top
  for (int i = CHUNK - 1; i >= 0; --i) {
    x = ch8[(i << 7) + x];
    pos[(i << 7) + threadIdx.x] = (unsigned char)x;
  }
  __syncthreads();
  int bt = btag[(c + 1) * B_N + b];         // true tag entering this chunk
  if (threadIdx.x < CHUNK)
    paths[(size_t)(c * CHUNK + threadIdx.x) * B_N + b] =
        pos[(threadIdx.x << 7) + bt];
}

// ===========================================================================
// Host-side launch. d_out = [paths int32 (512*256)] ++ [best_scores f32 (256)]
// ===========================================================================
extern "C" void kernel_launch(void* const* d_in, const int* in_sizes, int n_in,
                              void* d_out, int out_size, void* d_ws, size_t ws_size,
                              hipStream_t stream)
{
  (void)in_sizes; (void)n_in; (void)out_size; (void)ws_size;

  const float* feats   = (const float*)d_in[0];
  const float* trans   = (const float*)d_in[1];
  const float* start_t = (const float*)d_in[2];
  const float* end_t   = (const float*)d_in[3];

  int*   paths  = (int*)d_out;
  float* scores = (float*)d_out + (size_t)S_LEN * B_N;

  // workspace layout (~16.2 MiB total)
  unsigned char* bp = (unsigned char*)d_ws;                      // 511*256*128 u8
  unsigned char* Fc = bp + (size_t)(S_LEN - 1) * B_N * T_N;      // 7*256*128 u8
  int* btag = (int*)(Fc + (size_t)NCHUNK * B_N * T_N);           // 8*256 i32
  int* bl   = btag + (NCHUNK + 1) * B_N;                         // 256 i32

  viterbi_fwd<<<B_N, T_N, 0, stream>>>(feats, trans, start_t, end_t,
                                       bp, bl, scores, paths);
  viterbi_compose<<<NCHUNK * B_N, T_N, 0, stream>>>(bp, Fc);
  viterbi_boundary<<<1, B_N, 0, stream>>>(Fc, bl, btag);
  viterbi_fill<<<NCHUNK * B_N, T_N, 0, stream>>>(bp, btag, paths);
}